// ISTFTNetModule_31550829756853
// MI455X (gfx1250) — compile-verified
//
#include <hip/hip_runtime.h>
#include <math.h>

// ISTFT -> STFT(phase) -> ISTFT, N_FFT=16, HOP=4, PAD=8, fully fused.
// B=32, K=9, T=65536, Lwav=Lout=(T-1)*4=262140.

#define NFFT    16
#define HOPS    4
#define PADS    8
#define TFR     65536
#define NBATCH  32
#define LWAV    262140
#define LOUT    262140
#define SOUT    500          // output samples per workgroup
#define NWAVES  8
#define THREADS 256
#define S1TILES 9            // 144 stage-1 frames per WG
#define FSTRIDE 20           // padded LDS frame row stride (bank-conflict free, 16B aligned)

typedef float v2f __attribute__((ext_vector_type(2)));
typedef float v8f __attribute__((ext_vector_type(8)));

// cos(pi*m/8), exact table (period-16 twiddles). sin(pi*m/8) = COS16[(m+12)&15].
__device__ __constant__ float COS16[16] = {
    1.0f,                0.92387953251128674f,  0.70710678118654752f,  0.38268343236508977f,
    0.0f,               -0.38268343236508977f, -0.70710678118654752f, -0.92387953251128674f,
   -1.0f,               -0.92387953251128674f, -0.70710678118654752f, -0.38268343236508977f,
    0.0f,                0.38268343236508977f,  0.70710678118654752f,  0.92387953251128674f };

// irfft coefficients: x[n] = sum_k Re[k]*ICOS(k,n) - Im[k]*ISIN(k,n)
// ICOS(k,n) = (wk/16) cos(2pi k n/16), ISIN(k,n) = (wk/16) sin(2pi k n/16), wk = 1 for k=0,8 else 2.
__device__ __forceinline__ float icos_c(int k, int n) {
  if (k > 8) return 0.0f;
  float wk = (k == 0 || k == 8) ? 0.0625f : 0.125f;
  return wk * COS16[(k * n) & 15];
}
__device__ __forceinline__ float isin_c(int k, int n) {
  if (k > 8) return 0.0f;
  float wk = (k == 0 || k == 8) ? 0.0625f : 0.125f;
  return wk * COS16[((k * n) + 12) & 15];
}

// D(16x16,f32) = A(16x4,f32) * B(4x16,f32) + C   (wave32, f32-exact)
__device__ __forceinline__ v8f wmma4(v2f a, v2f b, v8f c) {
  return __builtin_amdgcn_wmma_f32_16x16x4_f32(false, a, false, b, (short)0, c, false, false);
}

__global__ __launch_bounds__(THREADS)
void istftnet_fused(const float* __restrict__ mag,
                    const float* __restrict__ win,
                    float* __restrict__ out)
{
  __shared__ float FR[S1TILES * 16 * FSTRIDE]; // frame tiles (reused by stage 1 then stage 2)
  __shared__ float wavt[528];                  // intermediate waveform tile (524 used)
  __shared__ float Ws[16];                     // hann window (from input, bit-faithful)

  const int tid  = threadIdx.x;
  const int lane = tid & 31;
  const int wv   = tid >> 5;       // wave id (uniform per wave)
  const int col  = lane & 15;      // matrix column / M index depending on operand
  const int hl   = lane >> 4;      // half-wave flag (WMMA operand layout)
  const int b    = blockIdx.y;
  const int j0   = blockIdx.x * SOUT;
  const int tA   = (j0 >> 2) - 1;  // first stage-2 frame this WG needs (128 frames)
  const int tS   = tA - 3;         // first stage-1 frame this WG needs (144 frames)
  const int wbase = 4 * tA - 8;    // first wav-signal sample this WG needs (524 samples)

  if (tid < 16) Ws[tid] = win[tid];

  const float* magb = mag + (size_t)b * 9 * TFR;

  // ---------------- stage 1: frames[t][n] = sum_k mag[k,t]*ICOS(k,n) via WMMA ----------------
  // D[M=n, N=t]; A[n,k]=ICOS(k,n) (constants, hoisted), B[k,t]=mag (K padded 9->12, 3 chunks).
  v2f a_s1[3];
#pragma unroll
  for (int c = 0; c < 3; ++c) {
    int k0 = c * 4 + hl * 2;
    a_s1[c].x = icos_c(k0, col);
    a_s1[c].y = icos_c(k0 + 1, col);
  }

  for (int tile = wv; tile < S1TILES; tile += NWAVES) {
    int t  = tS + tile * 16 + col;
    float mk = (t >= 0 && t < TFR) ? 1.0f : 0.0f;        // 0/1 mask instead of branchy loads
    int  tc  = t < 0 ? 0 : (t >= TFR ? TFR - 1 : t);     // clamped (always-valid) address
    const float* mp = magb + tc;
    v8f d = {};
#pragma unroll
    for (int c = 0; c < 3; ++c) {
      int k0 = c * 4 + hl * 2;
      int ra = k0     > 8 ? 8 : k0;
      int rb = k0 + 1 > 8 ? 8 : k0 + 1;
      v2f bb;
      bb.x = (k0     <= 8 ? mk : 0.0f) * mp[(size_t)ra * TFR];
      bb.y = (k0 + 1 <= 8 ? mk : 0.0f) * mp[(size_t)rb * TFR];
      d = wmma4(a_s1[c], bb, d);
    }
    int base = (tile * 16 + col) * FSTRIDE + hl * 8;  // lane holds frame t, samples n=8*hl..8*hl+7
#pragma unroll
    for (int r = 0; r < 8; ++r) FR[base + r] = d[r];
  }
  __syncthreads();

  // ---------------- overlap-add gather -> wav tile (deterministic, no atomics) ----------------
  for (int i = tid; i < 524; i += THREADS) {
    int jw = wbase + i;
    float v = 0.0f;
    if (jw >= 0 && jw < LWAV) {
      int l = jw + PADS;
      int thi = l >> 2, n0 = l & 3;
      float num = 0.0f, den = 0.0f;
#pragma unroll
      for (int m = 0; m < 4; ++m) {
        int t = thi - m;
        int n = n0 + 4 * m;
        float w  = Ws[n];
        float ok = ((t >= 0) && (t < TFR)) ? 1.0f : 0.0f;
        num += ok * w * FR[(t - tS) * FSTRIDE + n];      // frame index always in [0,144)
        den += ok * w * w;
      }
      v = num / den;
    }
    wavt[i] = v;
  }
  __syncthreads();

  // ---------------- stage 2: STFT -> unit phase -> mag remix -> inverse frames ----------------
  {
    int t2 = tA + wv * 16 + col;
    float mk2 = (t2 >= 0 && t2 < TFR) ? 1.0f : 0.0f;
    int  t2c  = t2 < 0 ? 0 : (t2 >= TFR ? TFR - 1 : t2);

    // 2a: Re[k,t']=sum_n xw[n]cos(2pi kn/16), Im[k,t']=-sum_n xw[n]sin(...)  (K=n=16 -> 4 chunks)
    v8f dre = {}, dim_ = {};
#pragma unroll
    for (int c = 0; c < 4; ++c) {
      int n0 = c * 4 + hl * 2;
      v2f bx, ac, as;
#pragma unroll
      for (int e = 0; e < 2; ++e) {
        int n = n0 + e;
        int p = 4 * t2c + n - PADS;            // reflect-padded signal coordinate (clamped frame)
        if (p < 0)      p = -p;
        if (p >= LWAV)  p = 2 * LWAV - 2 - p;
        int li = p - wbase;
        li = li < 0 ? 0 : (li > 523 ? 523 : li);
        float xw = mk2 * Ws[n] * wavt[li];
        float cc = (col <= 8) ?  COS16[(col * n) & 15]        : 0.0f;
        float ss = (col <= 8) ? -COS16[((col * n) + 12) & 15] : 0.0f;
        if (e == 0) { bx.x = xw; ac.x = cc; as.x = ss; }
        else        { bx.y = xw; ac.y = cc; as.y = ss; }
      }
      dre  = wmma4(ac, bx, dre);
      dim_ = wmma4(as, bx, dim_);
    }

    // pointwise: spec = mag * (Re,Im)/|S|;  atan2(0,0)=0 -> (1,0)
    float Re2[8], Im2[8];
#pragma unroll
    for (int r = 0; r < 8; ++r) {
      int k  = r + hl * 8;                     // D layout: lane holds (k, t')
      int kc = k > 8 ? 8 : k;                  // clamped row, masked below
      float re = dre[r], im = dim_[r];
      float rr = re * re + im * im;
      float inv = (rr > 0.0f) ? (1.0f / sqrtf(rr)) : 0.0f;
      float cph = (rr > 0.0f) ? re * inv : 1.0f;
      float sph = im * inv;                    // 0 when rr==0
      float okm = (r == 0) ? mk2 : (hl ? 0.0f : mk2);   // k<=8 mask folds per unrolled r
      float m = okm * magb[(size_t)kc * TFR + t2c];
      Re2[r] = m * cph;
      Im2[r] = m * sph;
    }

    // 2c: g[n,t'] = sum_k Re2*ICOS - Im2*ISIN. Re-layout D->B via cross-half shuffles.
    v8f dg = {};
#pragma unroll
    for (int c = 0; c < 2; ++c) {              // K chunks {0..3}, {4..7}
      int kb = 4 * c;
      float srx = __shfl_xor(Re2[kb + 2], 16);
      float sry = __shfl_xor(Re2[kb + 3], 16);
      float six = __shfl_xor(Im2[kb + 2], 16);
      float siy = __shfl_xor(Im2[kb + 3], 16);
      v2f br, bi, a1, a2;
      br.x = hl ? srx : Re2[kb + 0];  br.y = hl ? sry : Re2[kb + 1];
      bi.x = hl ? six : Im2[kb + 0];  bi.y = hl ? siy : Im2[kb + 1];
      int kk = kb + hl * 2;
      a1.x =  icos_c(kk, col);  a1.y =  icos_c(kk + 1, col);
      a2.x = -isin_c(kk, col);  a2.y = -isin_c(kk + 1, col);
      dg = wmma4(a1, br, dg);
      dg = wmma4(a2, bi, dg);
    }
    {                                          // K chunk {8..11}: only k=8 nonzero
      float srx = __shfl_xor(Re2[0], 16);      // low lanes fetch partner's k=8 value
      float six = __shfl_xor(Im2[0], 16);
      v2f br, bi, a1, a2;
      br.x = hl ? 0.0f : srx;  br.y = 0.0f;
      bi.x = hl ? 0.0f : six;  bi.y = 0.0f;
      int kk = 8 + hl * 2;
      a1.x =  icos_c(kk, col);  a1.y =  icos_c(kk + 1, col);
      a2.x = -isin_c(kk, col);  a2.y = -isin_c(kk + 1, col);
      dg = wmma4(a1, br, dg);
      dg = wmma4(a2, bi, dg);
    }

    int base = (wv * 16 + col) * FSTRIDE + hl * 8;   // FR reused: safe after barrier above
#pragma unroll
    for (int r = 0; r < 8; ++r) FR[base + r] = dg[r];
  }
  __syncthreads();

  // ---------------- final overlap-add gather -> output ----------------
  for (int i = tid; i < SOUT; i += THREADS) {
    int j = j0 + i;
    if (j < LOUT) {
      int l = j + PADS;
      int thi = l >> 2, n0 = l & 3;
      float num = 0.0f, den = 0.0f;
#pragma unroll
      for (int m = 0; m < 4; ++m) {
        int t = thi - m;
        int n = n0 + 4 * m;
        float w  = Ws[n];
        float ok = ((t >= 0) && (t < TFR)) ? 1.0f : 0.0f;
        num += ok * w * FR[(t - tA) * FSTRIDE + n];
        den += ok * w * w;
      }
      out[(size_t)b * LOUT + j] = num / den;
    }
  }
}

extern "C" void kernel_launch(void* const* d_in, const int* in_sizes, int n_in,
                              void* d_out, int out_size, void* d_ws, size_t ws_size,
                              hipStream_t stream) {
  (void)in_sizes; (void)n_in; (void)out_size; (void)d_ws; (void)ws_size;
  const float* mag = (const float*)d_in[0];   // (32, 9, 65536) f32
  const float* win = (const float*)d_in[1];   // (16,) f32 hann
  float* out = (float*)d_out;                 // (32, 262140) f32
  dim3 grid((LOUT + SOUT - 1) / SOUT, NBATCH);
  istftnet_fused<<<grid, dim3(THREADS), 0, stream>>>(mag, win, out);
}